// AutoAUG_163208757588
// MI455X (gfx1250) — compile-verified
//
#include <hip/hip_runtime.h>
#include <cmath>

#define TDIM 2048
#define CDIM 64
#define BDIM 64
#define LWIN 1844           // ceil(0.9*2048)
#define ROW8 8              // LUT row: base, coef[0..5], pad
#define LUT_FLOATS (TDIM * ROW8)   // 16384 floats = 64 KB

struct SplineConsts {
  float G[36];     // Ainv @ D  (maps y[0..5] -> M[0..5])
  float knots[6];
  float h;
};

// ---------------------------------------------------------------------------
// Kernel 1: per-timestep spline coefficients  S[b,c,t] = base[t] + coef[t][j]*wk[b,j,c]
// plus the softmax gate weights para[0][0..3].  Writes d_ws[0..16387].
// ---------------------------------------------------------------------------
__global__ __launch_bounds__(256) void coef_para_kernel(
    float* __restrict__ ws, const float* __restrict__ weight,
    const float* __restrict__ gate_eps, SplineConsts sc) {
  const int t = blockIdx.x * blockDim.x + threadIdx.x;
  if (t < TDIM) {
    const float h = sc.h;
    const float tf = (float)t;
    int seg = (int)(tf / h);
    seg = seg < 0 ? 0 : (seg > 4 ? 4 : seg);
    const float dl = tf - sc.knots[seg];
    const float dr = sc.knots[seg + 1] - tf;
    const float inv6h = 1.0f / (6.0f * h);
    // S = (M[s]*dr^3 + M[s+1]*dl^3)/(6h) + (y[s]/h - M[s]h/6)dr + (y[s+1]/h - M[s+1]h/6)dl
    const float p = dr * dr * dr * inv6h - h * dr * (1.0f / 6.0f);  // coeff of M[seg]
    const float q = dl * dl * dl * inv6h - h * dl * (1.0f / 6.0f);  // coeff of M[seg+1]
    float w[6];
#pragma unroll
    for (int j = 0; j < 6; ++j)
      w[j] = p * sc.G[seg * 6 + j] + q * sc.G[(seg + 1) * 6 + j];
    w[seg]     += dr / h;
    w[seg + 1] += dl / h;
    float base = 0.0f;
#pragma unroll
    for (int j = 0; j < 6; ++j) base += w[j] * sc.knots[j];   // y0[j] = knots[j]
    float* row = ws + t * ROW8;
    row[0] = base;
#pragma unroll
    for (int j = 0; j < 6; ++j) row[1 + j] = 0.2f * sc.knots[j] * w[j]; // dS/dwk
    row[7] = 0.0f;
  }
  if (blockIdx.x == 0 && threadIdx.x == 0) {
    // para[0] = softmax(logit(eps) + weight[0])  with eps = 0.9999 - 0.9998*ge
    float g[4], m = -1e30f;
    for (int k = 0; k < 4; ++k) {
      const float eps = 0.9999f - 0.9998f * gate_eps[k];      // row 0 of (2,4)
      const float gate = logf(eps) - log1pf(-eps);
      g[k] = gate + weight[k];                                 // row 0 of (2,4)
      m = fmaxf(m, g[k]);
    }
    float s = 0.0f;
    for (int k = 0; k < 4; ++k) { g[k] = expf(g[k] - m); s += g[k]; }
    const float inv = 1.0f / s;
    for (int k = 0; k < 4; ++k) ws[LUT_FLOATS + k] = g[k] * inv;
  }
}

// ---------------------------------------------------------------------------
// Lazy spline-position evaluation out of the LDS LUT (two ds_load_b128 + 6 FMA)
// ---------------------------------------------------------------------------
__device__ __forceinline__ float eval_xp(const float* lut, int idx,
                                         const float wk[6], float scale) {
  const float4 a = *reinterpret_cast<const float4*>(lut + (idx << 3));
  const float4 b = *reinterpret_cast<const float4*>(lut + (idx << 3) + 4);
  const float S = a.x + a.y * wk[0] + a.z * wk[1] + a.w * wk[2]
                + b.x * wk[3] + b.y * wk[4] + b.z * wk[5];
  return fminf(fmaxf(scale * S, 0.0f), 2047.0f);
}

// ---------------------------------------------------------------------------
// Kernel 2: fused augmentation.  Block = (c:64) x (4 t-subgroups of 32),
// 16 t-tiles per batch element -> 1024 blocks, 256 threads (8 wave32).
// LUT filled with global_load_async_to_lds_b128 (ASYNCcnt path).
// Search bracket (xl,xr,fl,fr) is carried incrementally: ~1 LDS eval and
// ~1 x-gather per output element in steady state.
// ---------------------------------------------------------------------------
__global__ __launch_bounds__(256) void aug_main_kernel(
    const float* __restrict__ x, const float* __restrict__ jit,
    const float* __restrict__ snoise, const float* __restrict__ wknots,
    const int* __restrict__ starts, const float* __restrict__ ws,
    float* __restrict__ out) {
  __shared__ __align__(16) float lut[LUT_FLOATS];  // 64 KB of the 320 KB WGP LDS
  const int tid = threadIdx.x;
  const int b = blockIdx.x >> 4;
  const int tile = blockIdx.x & 15;

#if defined(__HIP_DEVICE_COMPILE__)
  {  // async bulk copy ws -> LDS: 4096 x b128, 16 per thread
    const unsigned lds0 = (unsigned)(unsigned long long)&lut[0];
    const char* g0 = (const char*)ws;
#pragma unroll
    for (int i = 0; i < 16; ++i) {
      const int f4 = i * 256 + tid;
      const unsigned laddr = lds0 + (unsigned)(f4 * 16);
      const unsigned long long gaddr = (unsigned long long)(g0 + (size_t)f4 * 16);
      asm volatile("global_load_async_to_lds_b128 %0, %1, off"
                   :: "v"(laddr), "v"(gaddr) : "memory");
    }
    asm volatile("s_wait_asynccnt 0x0" ::: "memory");
  }
#else
  for (int i = tid; i < LUT_FLOATS; i += 256) lut[i] = ws[i];
#endif
  __syncthreads();

  const int c = tid & 63;
  const int sg = tid >> 6;
  const int t0 = tile * 128 + sg * 32;

  float wk[6];
#pragma unroll
  for (int j = 0; j < 6; ++j) wk[j] = wknots[(size_t)b * 384 + j * 64 + c];

  // scale = (T-1)/S[..,-1]
  const float* rl = lut + 2047 * ROW8;
  const float Slast = rl[0] + rl[1] * wk[0] + rl[2] * wk[1] + rl[3] * wk[2]
                    + rl[4] * wk[3] + rl[5] * wk[4] + rl[6] * wk[5];
  const float scale = 2047.0f / Slast;

  const float p0 = ws[LUT_FLOATS + 0];
  const float p1 = ws[LUT_FLOATS + 1];
  const float p2 = ws[LUT_FLOATS + 2];
  const float p3 = ws[LUT_FLOATS + 3];

  const int st = starts[b];
  const size_t boff = (size_t)b * (TDIM * CDIM);
  const float* xb = x + boff;
  const float* jb = jit + boff;
  float* o1 = out + boff;
  float* o2 = out + (size_t)BDIM * TDIM * CDIM + boff;

  // searchsorted(xp, q, 'right'): initial binary search for q = t0, then
  // monotone warm-start advance (t strictly increasing within the thread).
  int idx;
  {
    const float q = (float)t0;
    int lo = 0, hi = TDIM;
    while (lo < hi) {
      const int mid = (lo + hi) >> 1;
      if (eval_xp(lut, mid, wk, scale) <= q) lo = mid + 1; else hi = mid;
    }
    idx = lo < 1 ? 1 : (lo > TDIM - 1 ? TDIM - 1 : lo);
  }
  // carried bracket state: xl=xp[idx-1], xr=xp[idx], fl=x[idx-1], fr=x[idx]
  float xl = eval_xp(lut, idx - 1, wk, scale);
  float xr = eval_xp(lut, idx, wk, scale);
  float fl = xb[(idx - 1) * CDIM + c];
  float fr = xb[idx * CDIM + c];

  const float posScale = (float)LWIN / 2047.0f;

  for (int tt = 0; tt < 32; ++tt) {
    const int t = t0 + tt;
    const float q = (float)t;
    const int eoff = t * CDIM + c;
    const float xv = xb[eoff];
    const float jv = jb[eoff];
    const float sn = snoise[t];

    const float a_jit = fmaf(0.8f, jv, xv);
    const float a_scl = xv * fmaf(0.1f, sn, 1.0f);

    // --- time-warp: inverse-map via interp(t, xp, x), incremental bracket ---
    while (xr <= q && idx < TDIM - 1) {
      ++idx;
      xl = xr; fl = fr;
      xr = eval_xp(lut, idx, wk, scale);
      fr = xb[idx * CDIM + c];
    }
    const float dx = xr - xl;
    const float a_tw = (dx > 0.0f) ? fmaf((q - xl) / dx, fr - fl, fl) : fl;

    // --- window slice + uniform resample ---
    float pos = fminf(q * posScale, (float)(LWIN - 1));
    int i0 = (int)pos;
    if (i0 > LWIN - 2) i0 = LWIN - 2;
    const float frac = pos - (float)i0;
    const float w0 = xb[(st + i0) * CDIM + c];
    const float w1 = xb[(st + i0 + 1) * CDIM + c];
    const float a_ws = fmaf(frac, w1 - w0, w0);

    o1[eoff] = p0 * a_jit + p1 * a_scl + p2 * a_tw + p3 * a_ws;
    o2[eoff] = xv;  // aug2 = x
  }
}

// ---------------------------------------------------------------------------
extern "C" void kernel_launch(void* const* d_in, const int* in_sizes, int n_in,
                              void* d_out, int out_size, void* d_ws, size_t ws_size,
                              hipStream_t stream) {
  const float* x  = (const float*)d_in[0];
  const float* w  = (const float*)d_in[1];
  const float* jn = (const float*)d_in[2];
  const float* sn = (const float*)d_in[3];
  const float* wk = (const float*)d_in[4];
  const float* ge = (const float*)d_in[5];
  const int*   ss = (const int*)d_in[6];
  float* out = (float*)d_out;
  float* ws  = (float*)d_ws;

  // Host-side (deterministic, no HIP calls): not-a-knot tridiagonal inverse,
  // G = Ainv @ D where D is the scaled second-difference RHS operator.
  SplineConsts sc;
  {
    const double h = 2047.0 / 5.0;
    double A[6][6] = {{0}}, I[6][6] = {{0}};
    A[0][0] = 1; A[0][1] = -2; A[0][2] = 1;
    A[5][3] = 1; A[5][4] = -2; A[5][5] = 1;
    for (int i = 1; i < 5; ++i) { A[i][i-1] = h; A[i][i] = 4*h; A[i][i+1] = h; }
    for (int i = 0; i < 6; ++i) I[i][i] = 1.0;
    for (int col = 0; col < 6; ++col) {            // Gauss-Jordan, partial pivot
      int piv = col; double best = fabs(A[col][col]);
      for (int r = col + 1; r < 6; ++r)
        if (fabs(A[r][col]) > best) { best = fabs(A[r][col]); piv = r; }
      if (piv != col)
        for (int k = 0; k < 6; ++k) {
          double ta = A[piv][k]; A[piv][k] = A[col][k]; A[col][k] = ta;
          double ti = I[piv][k]; I[piv][k] = I[col][k]; I[col][k] = ti;
        }
      const double d = 1.0 / A[col][col];
      for (int k = 0; k < 6; ++k) { A[col][k] *= d; I[col][k] *= d; }
      for (int r = 0; r < 6; ++r) if (r != col) {
        const double f = A[r][col];
        if (f != 0.0)
          for (int k = 0; k < 6; ++k) { A[r][k] -= f * A[col][k]; I[r][k] -= f * I[col][k]; }
      }
    }
    double D[6][6] = {{0}};
    for (int i = 1; i < 5; ++i) { D[i][i-1] = 6.0/h; D[i][i] = -12.0/h; D[i][i+1] = 6.0/h; }
    for (int m = 0; m < 6; ++m)
      for (int j = 0; j < 6; ++j) {
        double s = 0.0;
        for (int n = 0; n < 6; ++n) s += I[m][n] * D[n][j];
        sc.G[m * 6 + j] = (float)s;
      }
    for (int j = 0; j < 6; ++j) sc.knots[j] = (float)(h * j);
    sc.h = (float)h;
  }

  coef_para_kernel<<<8, 256, 0, stream>>>(ws, w, ge, sc);
  aug_main_kernel<<<1024, 256, 0, stream>>>(x, jn, sn, wk, ss, ws, out);
}